// BinsChamferLoss_82403242541717
// MI455X (gfx1250) — compile-verified
//
#include <hip/hip_runtime.h>
#include <math.h>

#define N_BATCH   4
#define P_BINS    256          // bin centers per batch (edges = 257)
#define M_PTS     76800        // 240*320
#define TILE      512          // points per LDS tile (M_PTS = 150 * 512 exactly)
#define NTILES    (M_PTS / TILE)
#define WPB       64           // workgroups per batch
#define BLOCK     256
#define MIN_DEPTH 0.001f
#define LOSS_W    0.1f

// ---- CDNA5 async global->LDS path (guarded; fallback = plain loads) ----
#if defined(__AMDGCN__) && __has_builtin(__builtin_amdgcn_global_load_async_to_lds_b64)
#define USE_ASYNC_LDS 1
#else
#define USE_ASYNC_LDS 0
#endif

// b64 async op moves one v2i (8 bytes) per lane; builtin params are
// address-space-qualified pointers to v2i (AS1 = global, AS3 = LDS).
typedef int v2i __attribute__((vector_size(8)));
typedef __attribute__((address_space(1))) v2i gl_v2i;
typedef __attribute__((address_space(3))) v2i lds_v2i;

__device__ __forceinline__ void wait_async_le0() {
#if USE_ASYNC_LDS
#if __has_builtin(__builtin_amdgcn_s_wait_asynccnt)
    __builtin_amdgcn_s_wait_asynccnt(0);
#else
    asm volatile("s_wait_asynccnt 0" ::: "memory");
#endif
#endif
}
__device__ __forceinline__ void wait_async_le1() {
#if USE_ASYNC_LDS
#if __has_builtin(__builtin_amdgcn_s_wait_asynccnt)
    __builtin_amdgcn_s_wait_asynccnt(1);
#else
    asm volatile("s_wait_asynccnt 1" ::: "memory");
#endif
#endif
}

// ------------------------------------------------------------------
// Kernel 1: init per-bin running minimum to +inf (bit pattern)
// ------------------------------------------------------------------
__global__ void chamfer_init(unsigned* __restrict__ binMin) {
    int i = blockIdx.x * blockDim.x + threadIdx.x;
    if (i < N_BATCH * P_BINS) binMin[i] = 0x7F800000u;   // +inf
}

// ------------------------------------------------------------------
// Kernel 2: main pass. Grid = N_BATCH * WPB blocks of 256 threads.
// Thread t owns bin center t. Point tiles are DOUBLE-BUFFERED through
// the CDNA5 async global->LDS DMA: tile k+1 is in flight while every
// lane scans tile k (cham_x min) and binary-searches its own 2 points
// against the sorted centers (cham_y). ASYNCcnt <= 1 releases the
// older tile; the loop-end barrier protects buffer reuse.
// ------------------------------------------------------------------
__global__ void chamfer_main(const float* __restrict__ bins,
                             const float* __restrict__ target,
                             unsigned* __restrict__ binMin,
                             float* __restrict__ ysumP,
                             float* __restrict__ ycntP) {
    __shared__ float bc[P_BINS];
    __shared__ __align__(16) float tileBuf[2][TILE];
    __shared__ float redf[BLOCK];
    __shared__ float redc[BLOCK];

    const int n  = blockIdx.x / WPB;
    const int wg = blockIdx.x % WPB;
    const int t  = threadIdx.x;

    // bin centers (edges are sorted -> centers sorted ascending)
    const float* bn = bins + n * (P_BINS + 1);
    bc[t] = 0.5f * (bn[t] + bn[t + 1]);

    const float  myC  = bc[t];          // own write, no barrier needed yet
    float        lmin = __builtin_inff();
    float        ysum = 0.0f;
    float        ycnt = 0.0f;

    const float* tp = target + (size_t)n * M_PTS;

    // stage tile `idx` into LDS buffer `dst` (one b64 per lane -> one
    // async instruction per wave; 8 waves cover the 2KB tile)
    auto issue = [&](float* dst, int idx) {
        const float* src = tp + idx * TILE + 2 * t;
#if USE_ASYNC_LDS
        __builtin_amdgcn_global_load_async_to_lds_b64(
            (gl_v2i*)src, (lds_v2i*)(dst + 2 * t), 0, 0);
#else
        dst[2 * t]     = src[0];
        dst[2 * t + 1] = src[1];
#endif
    };

    issue(tileBuf[0], wg);                       // prologue: first tile
    __syncthreads();                             // also publishes bc[]

    int buf = 0;
    for (int idx = wg; idx < NTILES; idx += WPB, buf ^= 1) {
        const int nextIdx = idx + WPB;
        if (nextIdx < NTILES) {
            issue(tileBuf[buf ^ 1], nextIdx);    // overlap next DMA
            wait_async_le1();                    // oldest (current) tile done
        } else {
            wait_async_le0();
        }
        __syncthreads();                         // tile visible to all waves
        float* tile = tileBuf[buf];

        // ---- cham_y: this thread's two points, 8-step binary search ----
        #pragma unroll
        for (int k = 0; k < 2; ++k) {
            float x = tile[2 * t + k];
            if (x >= MIN_DEPTH) {
                int j;
                if (bc[P_BINS - 1] < x) {
                    j = P_BINS;                  // above all centers
                } else {
                    j = 0;                       // lower_bound in [0,255]
                    #pragma unroll
                    for (int s = 128; s > 0; s >>= 1) {
                        int c = j + s;
                        if (bc[c - 1] < x) j = c;
                    }
                }
                float dl = (j > 0)      ? (x - bc[j - 1]) : __builtin_inff();
                float dr = (j < P_BINS) ? (bc[j] - x)     : __builtin_inff();
                float dm = fminf(dl, dr);
                ysum += dm * dm;
                ycnt += 1.0f;
            } else {
                tile[2 * t + k] = 1e30f;  // sentinel: (myC-1e30)^2 -> +inf
            }
        }
        __syncthreads();

        // ---- cham_x: every lane scans the whole tile (LDS broadcast) ----
        const float4* t4 = (const float4*)tile;
        #pragma unroll 4
        for (int i = 0; i < TILE / 4; ++i) {
            float4 p = t4[i];
            float d0 = myC - p.x, d1 = myC - p.y;
            float d2 = myC - p.z, d3 = myC - p.w;
            lmin = fminf(lmin, fminf(fminf(d0 * d0, d1 * d1),
                                     fminf(d2 * d2, d3 * d3)));
        }
        __syncthreads();   // all waves done reading before DMA reuses buffer
    }

    // ---- per-block reduction of ysum / ycnt (deterministic tree) ----
    redf[t] = ysum;
    redc[t] = ycnt;
    __syncthreads();
    #pragma unroll
    for (int s = BLOCK / 2; s > 0; s >>= 1) {
        if (t < s) { redf[t] += redf[t + s]; redc[t] += redc[t + s]; }
        __syncthreads();
    }
    if (t == 0) {
        ysumP[blockIdx.x] = redf[0];
        ycntP[blockIdx.x] = redc[0];
    }

    // exact, order-independent global min (non-negative f32 bits == uint order)
    atomicMin(&binMin[n * P_BINS + t], __float_as_uint(lmin));
}

// ------------------------------------------------------------------
// Kernel 3: finalize. 1 block of 256 threads; fixed-order reductions.
// ------------------------------------------------------------------
__global__ void chamfer_finalize(const unsigned* __restrict__ binMin,
                                 const float* __restrict__ ysumP,
                                 const float* __restrict__ ycntP,
                                 float* __restrict__ out) {
    __shared__ float red[BLOCK];
    __shared__ float chamx[N_BATCH];
    __shared__ float ys[BLOCK];
    __shared__ float yc[BLOCK];
    const int t = threadIdx.x;

    // cham_x[n] = mean over bins of min distance^2
    for (int n = 0; n < N_BATCH; ++n) {
        red[t] = __uint_as_float(binMin[n * P_BINS + t]);
        __syncthreads();
        #pragma unroll
        for (int s = BLOCK / 2; s > 0; s >>= 1) {
            if (t < s) red[t] += red[t + s];
            __syncthreads();
        }
        if (t == 0) chamx[n] = red[0] * (1.0f / P_BINS);
        __syncthreads();
    }

    // cham_y partials: one partial per workgroup; N_BATCH*WPB == BLOCK
    ys[t] = ysumP[t];
    yc[t] = ycntP[t];
    __syncthreads();
    #pragma unroll
    for (int s = WPB / 2; s > 0; s >>= 1) {
        if ((t & (WPB - 1)) < s) { ys[t] += ys[t + s]; yc[t] += yc[t + s]; }
        __syncthreads();
    }

    if (t == 0) {
        float loss = 0.0f;
        for (int n = 0; n < N_BATCH; ++n)
            loss += chamx[n] + ys[n * WPB] / yc[n * WPB];
        out[0] = LOSS_W * (loss * (1.0f / N_BATCH));
    }
}

// ------------------------------------------------------------------
extern "C" void kernel_launch(void* const* d_in, const int* in_sizes, int n_in,
                              void* d_out, int out_size, void* d_ws, size_t ws_size,
                              hipStream_t stream) {
    (void)in_sizes; (void)n_in; (void)out_size; (void)ws_size;
    const float* bins   = (const float*)d_in[0];   // (4, 257) f32
    const float* target = (const float*)d_in[1];   // (4, 240, 320) f32
    float*       out    = (float*)d_out;           // scalar f32

    // workspace layout: [binMin: N*P u32][ysumP: N*WPB f32][ycntP: N*WPB f32]
    unsigned* binMin = (unsigned*)d_ws;
    float*    ysumP  = (float*)((char*)d_ws + (size_t)N_BATCH * P_BINS * sizeof(unsigned));
    float*    ycntP  = ysumP + N_BATCH * WPB;

    hipLaunchKernelGGL(chamfer_init,
                       dim3((N_BATCH * P_BINS + BLOCK - 1) / BLOCK), dim3(BLOCK),
                       0, stream, binMin);
    hipLaunchKernelGGL(chamfer_main,
                       dim3(N_BATCH * WPB), dim3(BLOCK),
                       0, stream, bins, target, binMin, ysumP, ycntP);
    hipLaunchKernelGGL(chamfer_finalize,
                       dim3(1), dim3(BLOCK),
                       0, stream, binMin, ysumP, ycntP, out);
}